// EMAQuantizer_77713138254467
// MI455X (gfx1250) — compile-verified
//
#include <hip/hip_runtime.h>
#include <math.h>

#define NUM_CODE   2048
#define CODE_DIM   64
#define N_TOKENS   131072
#define EMA_DECAY  0.99f

// LDS staging: 8 chunks of 256 codes, double buffered (2 x 64 KB = 128 KB LDS)
#define CHUNK_CODES   256
#define N_CHUNKS      (NUM_CODE / CHUNK_CODES)          // 8
#define CHUNK_FLOATS  (CHUNK_CODES * CODE_DIM)          // 16384
#define CHUNK_BYTES   (CHUNK_FLOATS * 4)                // 65536
#define CHUNK_TILES   (CHUNK_CODES / 16)                // 16

// Each wave handles 32 tokens (two 16-row M tiles); 8 waves per block.
#define TOK_PER_WAVE  32
#define WAVES_PER_BLK 8
#define TOK_PER_BLK   (TOK_PER_WAVE * WAVES_PER_BLK)    // 256

typedef __attribute__((ext_vector_type(2))) float v2f;
typedef __attribute__((ext_vector_type(8))) float v8f;
typedef __attribute__((ext_vector_type(4))) unsigned int v4u;
typedef __attribute__((ext_vector_type(4))) int v4i;
typedef __attribute__((ext_vector_type(8))) int v8i;

// ---------------------------------------------------------------------------
// Workspace (floats): [0,2048) cnorm | [2048,+131072) code_sum_batch
//                     | [133120,+2048) code_count_batch | [135168] sq_err
// Output (floats, tuple order): x_out | commit_loss | perplexity
//                               | new_codebook | new_sum | new_count
// ---------------------------------------------------------------------------
#define WS_CNORM   0
#define WS_SUMB    (NUM_CODE)
#define WS_CNTB    (NUM_CODE + NUM_CODE * CODE_DIM)
#define WS_ERR     (NUM_CODE + NUM_CODE * CODE_DIM + NUM_CODE)
#define WS_TOTAL   (WS_ERR + 1)

#define OUT_XOUT   0
#define OUT_LOSS   ((size_t)N_TOKENS * CODE_DIM)
#define OUT_PERP   (OUT_LOSS + 1)
#define OUT_NCB    (OUT_PERP + 1)
#define OUT_NSUM   (OUT_NCB + (size_t)NUM_CODE * CODE_DIM)
#define OUT_NCNT   (OUT_NSUM + (size_t)NUM_CODE * CODE_DIM)

#if defined(__has_builtin)
#if __has_builtin(__builtin_amdgcn_tensor_load_to_lds)
#define HAVE_TDM 1
#endif
#endif

// ---------- Kernel 1: code norms + zero the batch accumulators -------------
__global__ void vq_init(const float* __restrict__ cb, float* __restrict__ ws) {
    int i = blockIdx.x * blockDim.x + threadIdx.x;
    if (i >= WS_TOTAL) return;
    if (i < NUM_CODE) {
        const float* r = cb + (size_t)i * CODE_DIM;
        float s = 0.f;
        #pragma unroll
        for (int j = 0; j < CODE_DIM; ++j) s += r[j] * r[j];
        ws[WS_CNORM + i] = s;
    } else {
        ws[i] = 0.f;
    }
}

#ifdef HAVE_TDM
// Issue a TDM DMA: 2D tile (64 x rows, f32) from global codebook into LDS.
// D# per CDNA5 ISA §8.3-8.6: group0 = {flags, lds_addr, gaddr_lo, gaddr_hi|type},
// group1 = {data_size, dims, tile dims, dim0 stride}; groups 2/3 unused (2D).
__device__ __forceinline__ void tdm_load_chunk(const float* gsrc,
                                               unsigned lds_byte_addr,
                                               unsigned rows) {
    unsigned long long ga = (unsigned long long)(uintptr_t)gsrc;
    v4u g0;
    g0.x = 0x1u;                                   // count=1, user descriptor
    g0.y = lds_byte_addr;                          // LDS destination (bytes)
    g0.z = (unsigned)(ga & 0xFFFFFFFFu);           // global_addr[31:0]
    g0.w = (unsigned)(ga >> 32) | (2u << 30);      // global_addr[56:32] | type=2
    v8i g1;
    g1[0] = (int)(2u << 16);                       // data_size = 2 (4 bytes)
    g1[1] = (int)((unsigned)CODE_DIM << 16);       // tensor_dim0 = 64
    g1[2] = (int)(rows << 16);                     // tensor_dim1 (low 16)
    g1[3] = (int)((unsigned)CODE_DIM << 16);       // tile_dim0 = 64
    g1[4] = (int)rows;                             // tile_dim1
    g1[5] = CODE_DIM;                              // tensor_dim0_stride = 64
    g1[6] = 0;
    g1[7] = 0;
    v4i z4 = {0, 0, 0, 0};
#if __clang_major__ >= 23
    v8i z8 = {0, 0, 0, 0, 0, 0, 0, 0};
    __builtin_amdgcn_tensor_load_to_lds(g0, g1, z4, z4, z8, 0);
#else
    __builtin_amdgcn_tensor_load_to_lds(g0, g1, z4, z4, 0);
#endif
}
#endif

// ---------- Kernel 2: WMMA distance GEMM + fused argmin / gather / scatter --
__global__ void __launch_bounds__(256) vq_assign(
    const float* __restrict__ x, const float* __restrict__ cb,
    float* __restrict__ ws, float* __restrict__ out)
{
    extern __shared__ float lds[];                 // 2 * CHUNK_FLOATS

    const float* cnorm   = ws + WS_CNORM;
    float*       sumb    = ws + WS_SUMB;
    float*       cntb    = ws + WS_CNTB;
    float*       err_acc = ws + WS_ERR;

    const int lane = threadIdx.x & 31;
    const int wave = threadIdx.x >> 5;
    const int l16  = lane & 15;
    const int koff = (lane >> 4) * 2;              // K sub-slice of this half-wave
    const size_t tokbase = ((size_t)blockIdx.x * WAVES_PER_BLK + wave) * TOK_PER_WAVE;

    // --- Preload A fragments for two 16-row M tiles (ISA 32-bit A layout).
    const float* xr0 = x + (tokbase + l16) * CODE_DIM;
    const float* xr1 = x + (tokbase + 16 + l16) * CODE_DIM;
    v2f a0[16], a1[16];
    #pragma unroll
    for (int k = 0; k < 16; ++k) {
        a0[k] = *(const v2f*)(xr0 + 4 * k + koff);
        a1[k] = *(const v2f*)(xr1 + 4 * k + koff);
    }

    float best0[8], best1[8];
    int   bidx0[8], bidx1[8];
    #pragma unroll
    for (int r = 0; r < 8; ++r) {
        best0[r] = 3.4e38f; bidx0[r] = 0;
        best1[r] = 3.4e38f; bidx1[r] = 0;
    }

    const unsigned lds_base = (unsigned)(uintptr_t)(&lds[0]);

    // --- Stage chunk 0, then pipeline: wait chunk c, kick chunk c+1, compute.
#ifdef HAVE_TDM
    if (wave == 0)
        tdm_load_chunk(cb, lds_base, CHUNK_CODES);
#else
    for (int i = threadIdx.x; i < CHUNK_FLOATS / 4; i += 256)
        *(((float4*)lds) + i) = *(((const float4*)cb) + i);
#endif

    for (int c = 0; c < N_CHUNKS; ++c) {
        const int buf = c & 1;
#ifdef HAVE_TDM
        if (wave == 0)
            __builtin_amdgcn_s_wait_tensorcnt(0);  // chunk c landed in LDS
#endif
        __syncthreads();                           // publish chunk c to block
#ifdef HAVE_TDM
        if (wave == 0 && c + 1 < N_CHUNKS)
            tdm_load_chunk(cb + (size_t)(c + 1) * CHUNK_FLOATS,
                           lds_base + (unsigned)((c + 1) & 1) * CHUNK_BYTES,
                           CHUNK_CODES);
#else
        if (c + 1 < N_CHUNKS) {
            const float4* src = (const float4*)(cb + (size_t)(c + 1) * CHUNK_FLOATS);
            float4* dst = (float4*)(lds + (size_t)((c + 1) & 1) * CHUNK_FLOATS);
            for (int i = threadIdx.x; i < CHUNK_FLOATS / 4; i += 256)
                dst[i] = src[i];
        }
#endif
        const float* lcb = lds + (size_t)buf * CHUNK_FLOATS;

        for (int t = 0; t < CHUNK_TILES; ++t) {
            const int lcode = t * 16 + l16;                  // code within chunk
            const int code  = c * CHUNK_CODES + lcode;       // global code id
            const float* brow = lcb + (size_t)lcode * CODE_DIM;

            // Batch all 16 B fragments from LDS, then run both WMMA chains.
            v2f b[16];
            #pragma unroll
            for (int k = 0; k < 16; ++k)
                b[k] = *(const v2f*)(brow + 4 * k + koff);

            v8f acc0 = {0.f,0.f,0.f,0.f,0.f,0.f,0.f,0.f};
            v8f acc1 = {0.f,0.f,0.f,0.f,0.f,0.f,0.f,0.f};
            #pragma unroll
            for (int k = 0; k < 16; ++k) {
                acc0 = __builtin_amdgcn_wmma_f32_16x16x4_f32(
                    false, a0[k], false, b[k], (short)0, acc0, false, false);
                acc1 = __builtin_amdgcn_wmma_f32_16x16x4_f32(
                    false, a1[k], false, b[k], (short)0, acc1, false, false);
            }

            // key = ||c||^2 - 2 x.c  (||x||^2 is row-constant -> same argmin)
            const float cn = cnorm[code];
            #pragma unroll
            for (int r = 0; r < 8; ++r) {
                float k0 = cn - 2.f * acc0[r];
                float k1 = cn - 2.f * acc1[r];
                if (k0 < best0[r]) { best0[r] = k0; bidx0[r] = code; }
                if (k1 < best1[r]) { best1[r] = k1; bidx1[r] = code; }
            }
        }
        __syncthreads();   // everyone done reading buf before chunk c+2 reuses it
    }

    // --- Argmin across the 16 lanes sharing each D-matrix row.
    #pragma unroll
    for (int r = 0; r < 8; ++r) {
        #pragma unroll
        for (int off = 1; off < 16; off <<= 1) {
            float ov = __shfl_xor(best0[r], off, 16);
            int   oi = __shfl_xor(bidx0[r], off, 16);
            if (ov < best0[r] || (ov == best0[r] && oi < bidx0[r])) {
                best0[r] = ov; bidx0[r] = oi;
            }
            float ow = __shfl_xor(best1[r], off, 16);
            int   oj = __shfl_xor(bidx1[r], off, 16);
            if (ow < best1[r] || (ow == best1[r] && oj < bidx1[r])) {
                best1[r] = ow; bidx1[r] = oj;
            }
        }
    }

    // --- Epilogue: x_out gather, commit loss, EMA segment-sum atomics.
    float err = 0.f;
    const int j = lane * 2;   // 32 lanes x 2 floats cover one 64-dim row
    #pragma unroll
    for (int r = 0; r < 8; ++r) {
        const int cidx[4] = { __shfl(bidx0[r], 0, 32), __shfl(bidx0[r], 16, 32),
                              __shfl(bidx1[r], 0, 32), __shfl(bidx1[r], 16, 32) };
        const size_t mrow[4] = { tokbase + r,      tokbase + 8 + r,
                                 tokbase + 16 + r, tokbase + 24 + r };
        #pragma unroll
        for (int q = 0; q < 4; ++q) {
            const int cc = cidx[q];
            const size_t m = mrow[q];
            v2f cv = *(const v2f*)(cb + (size_t)cc * CODE_DIM + j);
            v2f xv = *(const v2f*)(x + m * CODE_DIM + j);
            *(v2f*)(out + OUT_XOUT + m * CODE_DIM + j) = cv;
            float dx = xv[0] - cv[0], dy = xv[1] - cv[1];
            err += dx * dx + dy * dy;
            atomicAdd(&sumb[cc * CODE_DIM + j],     xv[0]);
            atomicAdd(&sumb[cc * CODE_DIM + j + 1], xv[1]);
            if (lane == 0) atomicAdd(&cntb[cc], 1.f);
        }
    }

    #pragma unroll
    for (int off = 16; off >= 1; off >>= 1)
        err += __shfl_xor(err, off, 32);
    if (lane == 0) atomicAdd(err_acc, err);
}

// ---------- Kernel 3: EMA update + codebook refresh ------------------------
__global__ void vq_finalize(const float* __restrict__ x,
                            const float* __restrict__ code_sum,
                            const float* __restrict__ code_count,
                            const float* __restrict__ ws,
                            float* __restrict__ out)
{
    const int c = blockIdx.x;
    const int d = threadIdx.x;
    const float* sumb = ws + WS_SUMB;
    const float* cntb = ws + WS_CNTB;

    float ncnt = EMA_DECAY * code_count[c] + (1.f - EMA_DECAY) * cntb[c];
    float ns   = EMA_DECAY * code_sum[(size_t)c * CODE_DIM + d]
               + (1.f - EMA_DECAY) * sumb[(size_t)c * CODE_DIM + d];
    float used = (ncnt >= 1.f) ? 1.f : 0.f;
    float refreshed = ns / fmaxf(ncnt, 1e-8f);
    float crand = x[(size_t)c * CODE_DIM + d];

    out[OUT_NCB  + (size_t)c * CODE_DIM + d] = used * refreshed + (1.f - used) * crand;
    out[OUT_NSUM + (size_t)c * CODE_DIM + d] = ns;
    if (d == 0) out[OUT_NCNT + c] = ncnt;
}

// ---------- Kernel 4: perplexity + commit loss scalars ---------------------
__global__ void vq_scalars(const float* __restrict__ ws, float* __restrict__ out) {
    __shared__ float red[256];
    const float* cntb = ws + WS_CNTB;
    float s = 0.f;
    for (int c = threadIdx.x; c < NUM_CODE; c += 256) {
        float p = cntb[c] * (1.f / (float)N_TOKENS);
        s += p * logf(p + 1e-7f);
    }
    red[threadIdx.x] = s;
    __syncthreads();
    for (int o = 128; o > 0; o >>= 1) {
        if (threadIdx.x < o) red[threadIdx.x] += red[threadIdx.x + o];
        __syncthreads();
    }
    if (threadIdx.x == 0) {
        out[OUT_LOSS] = ws[WS_ERR] / (float)((size_t)N_TOKENS * CODE_DIM);
        out[OUT_PERP] = expf(-red[0]);
    }
}

// ---------------------------------------------------------------------------
extern "C" void kernel_launch(void* const* d_in, const int* in_sizes, int n_in,
                              void* d_out, int out_size, void* d_ws, size_t ws_size,
                              hipStream_t stream) {
    const float* x          = (const float*)d_in[0];
    const float* codebook   = (const float*)d_in[1];
    const float* code_sum   = (const float*)d_in[2];
    const float* code_count = (const float*)d_in[3];
    float* out = (float*)d_out;
    float* ws  = (float*)d_ws;

    vq_init<<<(WS_TOTAL + 255) / 256, 256, 0, stream>>>(codebook, ws);
    vq_assign<<<N_TOKENS / TOK_PER_BLK, 256, 2 * CHUNK_BYTES, stream>>>(
        x, codebook, ws, out);
    vq_finalize<<<NUM_CODE, CODE_DIM, 0, stream>>>(x, code_sum, code_count, ws, out);
    vq_scalars<<<1, 256, 0, stream>>>(ws, out);
}